// GraphNorm_42417097015723
// MI455X (gfx1250) — compile-verified
//
#include <hip/hip_runtime.h>
#include <hip/hip_bf16.h>

typedef float v2f __attribute__((ext_vector_type(2)));
typedef float v8f __attribute__((ext_vector_type(8)));

#define TILE_ROWS 256
#define LDS_STRIDE 136   // 128 + 8 pad floats: rows r and r+2 hit disjoint LDS banks; 544B = 16B-aligned
#define FEAT 128
#define NTHREADS 256
#define GN_EPS 1e-6f

__device__ __forceinline__ int lower_bound_i32(const int* __restrict__ b, int n, int v) {
    int lo = 0, hi = n;
    while (lo < hi) { int m = (lo + hi) >> 1; if (b[m] < v) lo = m + 1; else hi = m; }
    return lo;
}

extern "C" __global__ void __launch_bounds__(NTHREADS)
graphnorm_kernel(const float* __restrict__ x,
                 const int* __restrict__ batch,
                 const int* __restrict__ gseg,
                 const float* __restrict__ weight,
                 const float* __restrict__ bias,
                 const float* __restrict__ mean_scale,
                 float* __restrict__ out,
                 int N)
{
    extern __shared__ float lds[];              // dynamic LDS, hardware offset 0
    float* smu = lds + TILE_ROWS * LDS_STRIDE;  // [128] mean * mean_scale
    float* ssc = smu + FEAT;                    // [128] weight * rstd
    float* sbi = ssc + FEAT;                    // [128] bias

    const int tid  = threadIdx.x;
    const int lane = tid & 31;
    const int lh   = lane & 15;
    const int rsel = (lane >> 4) << 1;          // 0 for lanes 0-15, 2 for lanes 16-31
    const int wcol = (tid >> 5) << 4;           // 16 columns per wave, 8 waves = 128
    const int G = gseg[0];

    for (int g = (int)blockIdx.x; g < G; g += (int)gridDim.x) {
        int start = lower_bound_i32(batch, N, g);
        int end   = lower_bound_i32(batch, N, g + 1);
        start = __builtin_amdgcn_readfirstlane(start);
        end   = __builtin_amdgcn_readfirstlane(end);
        const int count = end - start;
        if (count <= 0) continue;               // uniform branch; no output rows exist

        const int ntiles = (count + TILE_ROWS - 1) / TILE_ROWS;

        v2f a; a.x = 1.0f; a.y = 1.0f;          // A = ones(16x4): layout-independent
        v8f csum = {};                          // accumulates column sums (all M rows equal)
        v8f csq  = {};                          // accumulates column sums of squares

        for (int t = 0; t < ntiles; ++t) {
            __syncthreads();                    // protect LDS reuse across tiles/segments
            const int r0 = t * TILE_ROWS;
            const int tr = (count - r0 < TILE_ROWS) ? (count - r0) : TILE_ROWS;
            const int rows_pad = (tr + 3) & ~3;

            // ---- async stage tile into LDS: 16B chunks, coalesced, ASYNCcnt-tracked ----
            const float* gbase = x + (size_t)(start + r0) * FEAT;
            const int nchunks = tr * (FEAT / 4);
            for (int q = tid; q < nchunks; q += NTHREADS) {
                const int row  = q >> 5;               // 32 chunks per 512B row
                const int c4   = q & 31;
                const int loff = row * (LDS_STRIDE * 4) + c4 * 16;  // LDS byte offset
                const float* ga = gbase + (size_t)q * 4;            // rows contiguous in global
                asm volatile("global_load_async_to_lds_b128 %0, %1, off"
                             :: "v"(loff), "v"(ga) : "memory");
            }
            // zero K-remainder rows so they contribute nothing to the WMMA sums
            for (int q = tid; q < (rows_pad - tr) * FEAT; q += NTHREADS) {
                lds[(tr + (q >> 7)) * LDS_STRIDE + (q & (FEAT - 1))] = 0.0f;
            }
            asm volatile("s_wait_asynccnt 0x0" ::: "memory");
            __syncthreads();

            // ---- column sums via v_wmma_f32_16x16x4_f32: D = ones(16x4) x X(4x16) + C ----
            // B layout (4x16 f32): v0 = {K=0 | K=2}, v1 = {K=1 | K=3} across lane halves.
            for (int r = 0; r < rows_pad; r += 4) {
                const float* p = lds + (r + rsel) * LDS_STRIDE + wcol + lh;
                const float b0 = p[0];
                const float b1 = p[LDS_STRIDE];
                v2f bv;  bv.x  = b0;      bv.y  = b1;
                v2f bsq; bsq.x = b0 * b0; bsq.y = b1 * b1;
                csum = __builtin_amdgcn_wmma_f32_16x16x4_f32(false, a, false, bv,
                                                             (short)0, csum, false, false);
                csq  = __builtin_amdgcn_wmma_f32_16x16x4_f32(false, a, false, bsq,
                                                             (short)0, csq,  false, false);
            }
        }

        // ---- per-column stats: every lane holds the sums for col = wcol + (lane&15) ----
        {
            const int col   = wcol + lh;
            const float rc   = 1.0f / (float)count;   // single divide, reused
            const float mean = csum[0] * rc;
            const float mu   = mean * mean_scale[col]; // sub = x - mu
            const float var  = csq[0] * rc - 2.0f * mu * mean + mu * mu;
            const float rstd = rsqrtf(var + GN_EPS);
            if (lane < 16) {                           // lanes 16-31 hold duplicates (M=8)
                smu[col] = mu;
                ssc[col] = weight[col] * rstd;
                sbi[col] = bias[col];
            }
        }
        __syncthreads();

        // ---- normalize in float4 chunks: chunk idx <-> global float offset 4*idx ----
        // idx = tid + k*NTHREADS keeps (tid & 31) fixed -> per-thread column block is
        // loop-invariant, so stats load once; one ds_load_b128 + global_store_b128 per 16B.
        {
            const int c4 = (tid & 31) << 2;            // first of 4 columns owned
            const float4 mu4 = *(const float4*)(smu + c4);
            const float4 sc4 = *(const float4*)(ssc + c4);
            const float4 bi4 = *(const float4*)(sbi + c4);
            float*       ob  = out + (size_t)start * FEAT;
            const int    ncnk = count * (FEAT / 4);

            if (ntiles == 1) {
                for (int idx = tid; idx < ncnk; idx += NTHREADS) {
                    const int r = idx >> 5;
                    const float4 v = *(const float4*)(lds + r * LDS_STRIDE + c4);
                    float4 o;
                    o.x = sc4.x * (v.x - mu4.x) + bi4.x;
                    o.y = sc4.y * (v.y - mu4.y) + bi4.y;
                    o.z = sc4.z * (v.z - mu4.z) + bi4.z;
                    o.w = sc4.w * (v.w - mu4.w) + bi4.w;
                    *(float4*)(ob + (size_t)idx * 4) = o;
                }
            } else {                                   // rare: segment spilled the tile
                const float* xb = x + (size_t)start * FEAT;
                for (int idx = tid; idx < ncnk; idx += NTHREADS) {
                    const float4 v = *(const float4*)(xb + (size_t)idx * 4);
                    float4 o;
                    o.x = sc4.x * (v.x - mu4.x) + bi4.x;
                    o.y = sc4.y * (v.y - mu4.y) + bi4.y;
                    o.z = sc4.z * (v.z - mu4.z) + bi4.z;
                    o.w = sc4.w * (v.w - mu4.w) + bi4.w;
                    *(float4*)(ob + (size_t)idx * 4) = o;
                }
            }
        }
    }
}

extern "C" void kernel_launch(void* const* d_in, const int* in_sizes, int n_in,
                              void* d_out, int out_size, void* d_ws, size_t ws_size,
                              hipStream_t stream) {
    const float* x          = (const float*)d_in[0];
    const int*   batch      = (const int*)d_in[1];
    const int*   g          = (const int*)d_in[2];   // num_segments (device scalar)
    const float* weight     = (const float*)d_in[3];
    const float* bias       = (const float*)d_in[4];
    const float* mean_scale = (const float*)d_in[5];
    float* out = (float*)d_out;
    const int N = in_sizes[1];

    const size_t shmem = (size_t)(TILE_ROWS * LDS_STRIDE + 3 * FEAT) * sizeof(float); // ~140.8 KB
    (void)hipFuncSetAttribute(reinterpret_cast<const void*>(graphnorm_kernel),
                              hipFuncAttributeMaxDynamicSharedMemorySize, (int)shmem);
    graphnorm_kernel<<<4096, NTHREADS, shmem, stream>>>(x, batch, g, weight, bias,
                                                        mean_scale, out, N);
}